// SpanConsistencyLayer_19026705121549
// MI455X (gfx1250) — compile-verified
//
#include <hip/hip_runtime.h>
#include <stdint.h>

// ---------------------------------------------------------------------------
// SpanConsistencyLayer for MI455X (gfx1250).
// B=512 rows, N=8192 tokens/row, span widths 2..15.
// boost[p] = max(masked[p], max_{spans of width 2..15 containing p}
//                               ww[w-2] * min(masked over span))
// out = scores + gamma * boost
//
// Memory-bound local stencil (29-wide window). CDNA5 paths used:
//   * GLOBAL_LOAD_ASYNC_TO_LDS_B128  (async DMA global->LDS, ASYNCcnt)
//   * s_wait_asynccnt                (split wait counter)
//   * LDS-resident stencil compute (ds_load_b128 + v_min/v_max VALU)
//   * b128 vectorized output stores
// ---------------------------------------------------------------------------

#define NEGV   (-1.0e9f)
#define SENT   (-3.0e38f)   // "minus infinity" sentinel: excludes invalid spans
#define TILE   2048
#define HALO   16           // >= 14, multiple of 4 so edge chunks stay whole
#define EXT    (TILE + 2*HALO)   // 2080 floats
#define NTHR   256
#define NROW   8192
#define TPR    (NROW / TILE)     // tiles per row = 4

// Low 32 bits of the generic pointer to LDS == workgroup-relative LDS byte
// address (generic LDS addresses are {aperture, offset[31:0]}).
__device__ __forceinline__ uint32_t lds_addr32(const void* p) {
    return (uint32_t)(uintptr_t)p;
}

// Async DMA 16 bytes global -> LDS. GV addressing: 64-bit VGPR address.
__device__ __forceinline__ void async_copy_b128(uint32_t lds_byte, const void* gptr) {
    asm volatile("global_load_async_to_lds_b128 %0, %1, off"
                 :: "v"(lds_byte), "v"(gptr)
                 : "memory");
}

__device__ __forceinline__ void wait_async_all() {
    asm volatile("s_wait_asynccnt 0" ::: "memory");
}

__global__ __launch_bounds__(NTHR)
void span_boost_kernel(const float* __restrict__ scores,
                       const int*   __restrict__ mask,
                       const float* __restrict__ gamma_p,
                       const float* __restrict__ wlogits,
                       float* __restrict__ out)
{
    __shared__ float sS[EXT];   // raw scores (extended tile)
    __shared__ int   sM[EXT];   // attention mask (extended tile)
    __shared__ float sX[EXT];   // masked scores w/ boundary sentinel
    __shared__ float sWW[16];   // softmax(width_logits)[0..13], [15]=gamma

    const int row   = blockIdx.x / TPR;
    const int tile  = blockIdx.x % TPR;
    const int start = tile * TILE;
    const int ext_start = start - HALO;               // may be negative

    const float* rowS = scores + (size_t)row * NROW;
    const int*   rowM = mask   + (size_t)row * NROW;

    // ---- stage extended tile into LDS via async DMA (16B chunks) ----------
    for (int c = threadIdx.x; c < EXT / 4; c += NTHR) {
        int g0 = ext_start + 4 * c;                   // chunk-aligned
        int gc = g0 < 0 ? 0 : (g0 > NROW - 4 ? NROW - 4 : g0);   // clamp
        async_copy_b128(lds_addr32(&sS[4 * c]), rowS + gc);
        async_copy_b128(lds_addr32(&sM[4 * c]), rowM + gc);
    }
    wait_async_all();
    __syncthreads();

    // ---- masking + boundary sentinel (4 elements / thread / iter) ---------
    for (int c = threadIdx.x; c < EXT / 4; c += NTHR) {
        const int4   m4 = *(const int4*)  &sM[4 * c];
        const float4 s4 = *(const float4*)&sS[4 * c];
        const int    g0 = ext_start + 4 * c;
        float4 x4;
        x4.x = (g0 + 0 < 0 || g0 + 0 >= NROW) ? SENT : (m4.x == 0 ? NEGV : s4.x);
        x4.y = (g0 + 1 < 0 || g0 + 1 >= NROW) ? SENT : (m4.y == 0 ? NEGV : s4.y);
        x4.z = (g0 + 2 < 0 || g0 + 2 >= NROW) ? SENT : (m4.z == 0 ? NEGV : s4.z);
        x4.w = (g0 + 3 < 0 || g0 + 3 >= NROW) ? SENT : (m4.w == 0 ? NEGV : s4.w);
        *(float4*)&sX[4 * c] = x4;
    }

    // ---- softmax over the 15 width logits (only [0..13] are consumed) -----
    if (threadIdx.x == 0) {
        float m = -3.0e38f;
        float e[15];
        #pragma unroll
        for (int i = 0; i < 15; ++i) m = fmaxf(m, wlogits[i]);
        float s = 0.0f;
        #pragma unroll
        for (int i = 0; i < 15; ++i) { e[i] = __expf(wlogits[i] - m); s += e[i]; }
        float inv = 1.0f / s;
        #pragma unroll
        for (int i = 0; i < 14; ++i) sWW[i] = e[i] * inv;
        sWW[15] = gamma_p[0];
    }
    __syncthreads();

    float ww[14];
    #pragma unroll
    for (int i = 0; i < 14; ++i) ww[i] = sWW[i];
    const float gamma = sWW[15];

    // ---- stencil: 4 consecutive tokens per thread share one 32-elt window -
    for (int t0 = 4 * (int)threadIdx.x; t0 < TILE; t0 += 4 * NTHR) {
        // window covers sX[t0+HALO-14 .. t0+HALO+17]  (32 floats)
        float Wn[32];
        #pragma unroll
        for (int i = 0; i < 32; ++i) Wn[i] = sX[t0 + HALO - 14 + i];

        const float4 s4 = *(const float4*)&sS[t0 + HALO];   // raw scores, aligned
        const float  sc[4] = { s4.x, s4.y, s4.z, s4.w };

        float4 r4;
        float res[4];
        #pragma unroll
        for (int q = 0; q < 4; ++q) {
            const int c = 14 + q;                 // token center within window
            float L[15], R[15];
            L[0] = Wn[c]; R[0] = Wn[c];
            #pragma unroll
            for (int i = 1; i < 15; ++i) {
                L[i] = fminf(L[i - 1], Wn[c - i]);   // min over [p-i .. p]
                R[i] = fminf(R[i - 1], Wn[c + i]);   // min over [p .. p+i]
            }
            float best = Wn[c];                    // max(boost, masked)
            #pragma unroll
            for (int w = 2; w <= 15; ++w) {
                float mw = SENT;                   // max over spans of width w
                #pragma unroll
                for (int k = 0; k < w; ++k)        // span [p-k, p+(w-1-k)]
                    mw = fmaxf(mw, fminf(L[k], R[w - 1 - k]));
                best = fmaxf(best, ww[w - 2] * mw);
            }
            res[q] = sc[q] + gamma * best;
        }
        r4.x = res[0]; r4.y = res[1]; r4.z = res[2]; r4.w = res[3];
        *(float4*)&out[(size_t)row * NROW + start + t0] = r4;   // b128 store
    }
}

extern "C" void kernel_launch(void* const* d_in, const int* in_sizes, int n_in,
                              void* d_out, int out_size, void* d_ws, size_t ws_size,
                              hipStream_t stream)
{
    const float* scores = (const float*)d_in[0];   // [B, N] f32
    const int*   amask  = (const int*)  d_in[1];   // [B, N] i32
    const float* gamma  = (const float*)d_in[2];   // scalar f32
    const float* wl     = (const float*)d_in[3];   // [15] f32
    float* out = (float*)d_out;

    const int B = in_sizes[0] / NROW;              // 512
    dim3 grid(B * TPR);                            // one block per 2048-token tile
    span_boost_kernel<<<grid, NTHR, 0, stream>>>(scores, amask, gamma, wl, out);
}